// graph_constructor_592705487497
// MI455X (gfx1250) — compile-verified
//
#include <hip/hip_runtime.h>
#include <math.h>

typedef __attribute__((ext_vector_type(2))) float v2f;
typedef __attribute__((ext_vector_type(8))) float v8f;

#define ALPHA 3.0f
#define NN 4096
#define DD 40

// ---------------------------------------------------------------------------
// Kernel 1: n1 = tanh(ALPHA*(emb1[idx] @ W1 + b1)), same for n2.
// One thread per (row, d) output element; 40-long dot products.
// ---------------------------------------------------------------------------
__global__ __launch_bounds__(256) void embed_kernel(
    const int* __restrict__ idx,
    const float* __restrict__ emb1, const float* __restrict__ emb2,
    const float* __restrict__ W1,   const float* __restrict__ b1,
    const float* __restrict__ W2,   const float* __restrict__ b2,
    float* __restrict__ n1, float* __restrict__ n2) {
  int tid = blockIdx.x * blockDim.x + threadIdx.x;
  if (tid >= NN * DD) return;
  int i = tid / DD;
  int d = tid - i * DD;
  int r = idx[i];
  const float* e1 = emb1 + (size_t)r * DD;
  const float* e2 = emb2 + (size_t)r * DD;
  float acc1 = b1[d];
  float acc2 = b2[d];
  #pragma unroll
  for (int k = 0; k < DD; ++k) {
    acc1 = fmaf(e1[k], W1[k * DD + d], acc1);
    acc2 = fmaf(e2[k], W2[k * DD + d], acc2);
  }
  n1[tid] = tanhf(ALPHA * acc1);
  n2[tid] = tanhf(ALPHA * acc2);
}

// ---------------------------------------------------------------------------
// Kernel 2: adj = relu(tanh(ALPHA*(n1@n2^T - n2@n1^T))), one wave per 16x16
// tile using V_WMMA_F32_16X16X4_F32 (f32 in/out; K=40 -> 10 chunks of 4).
// f32 WMMA has no A/B negate modifier, so keep two accumulators & subtract.
//
// Fragment layouts (wave32):
//   A (16x4, MxK): lanes 0-15 M=0..15 {VGPR0:K=k0, VGPR1:K=k0+1};
//                  lanes 16-31 M=0..15 {VGPR0:K=k0+2, VGPR1:K=k0+3}.
//   B (4x16, KxN) with B = X^T (X row-major [N,D]) needs B[k][n] = X[n][k],
//   which lands on the identical per-lane pattern: X[(base+m)*D + k0+2*half +{0,1}].
//   C/D (16x16): lane<16 -> (M=v, N=lane); lane>=16 -> (M=v+8, N=lane-16).
// ---------------------------------------------------------------------------
__global__ __launch_bounds__(32) void adj_wmma_kernel(
    const float* __restrict__ n1, const float* __restrict__ n2,
    float* __restrict__ out) {
  const int lane = threadIdx.x;
  const int m    = lane & 15;
  const int half = lane >> 4;
  const int rowBase = blockIdx.y * 16;
  const int colBase = blockIdx.x * 16;

  const float* n1r = n1 + (size_t)(rowBase + m) * DD;  // A1 rows (n1_i)
  const float* n2r = n2 + (size_t)(rowBase + m) * DD;  // A2 rows (n2_i)
  const float* n2c = n2 + (size_t)(colBase + m) * DD;  // B2 cols (n2_j^T)
  const float* n1c = n1 + (size_t)(colBase + m) * DD;  // B1 cols (n1_j^T)

  v8f c1 = {};  // n1_i . n2_j
  v8f c2 = {};  // n2_i . n1_j
  #pragma unroll
  for (int k0 = 0; k0 < DD; k0 += 4) {
    const int kk = k0 + 2 * half;
    v2f a1; a1.x = n1r[kk]; a1.y = n1r[kk + 1];
    v2f b2; b2.x = n2c[kk]; b2.y = n2c[kk + 1];
    v2f a2; a2.x = n2r[kk]; a2.y = n2r[kk + 1];
    v2f b1; b1.x = n1c[kk]; b1.y = n1c[kk + 1];
    c1 = __builtin_amdgcn_wmma_f32_16x16x4_f32(false, a1, false, b2,
                                               (short)0, c1, false, false);
    c2 = __builtin_amdgcn_wmma_f32_16x16x4_f32(false, a2, false, b1,
                                               (short)0, c2, false, false);
  }

  #pragma unroll
  for (int v = 0; v < 8; ++v) {
    const int row = rowBase + v + 8 * half;
    const int col = colBase + m;
    const float a = c1[v] - c2[v];
    const float adj = fmaxf(tanhf(ALPHA * a), 0.0f);
    out[(size_t)row * NN + col] = adj;
  }
}

// ---------------------------------------------------------------------------
// Kernel 3: per-row top-k on (adj + 0.01*noise); keep only selected adj
// entries, zero the rest. One 256-thread block per row. Iterative block-wide
// argmax (k rounds), lowest-index tie-break to match jax.lax.top_k.
// LDS: 16KB scores + small reduction/bitmask arrays.
// ---------------------------------------------------------------------------
__global__ __launch_bounds__(256) void topk_kernel(
    float* __restrict__ out, const float* __restrict__ noise,
    const int* __restrict__ kptr) {
  __shared__ float    score[NN];
  __shared__ float    red_val[256];
  __shared__ int      red_idx[256];
  __shared__ unsigned selmask[NN / 32];

  const int row = blockIdx.x;
  const int t   = threadIdx.x;
  int k = kptr[0];
  if (k > NN) k = NN;

  float*       orow = out   + (size_t)row * NN;
  const float* nrow = noise + (size_t)row * NN;

  for (int j = t; j < NN; j += 256) {
    score[j] = orow[j] + 0.01f * nrow[j];
  }
  for (int j = t; j < NN / 32; j += 256) selmask[j] = 0u;
  __syncthreads();

  for (int r = 0; r < k; ++r) {
    // thread-local argmax over strided slice (16 elements each)
    float best = -INFINITY;
    int   bidx = NN;
    for (int j = t; j < NN; j += 256) {
      const float s = score[j];
      if (s > best) { best = s; bidx = j; }
    }
    red_val[t] = best;
    red_idx[t] = bidx;
    __syncthreads();
    // block tree reduction; prefer larger value, then smaller index
    for (int off = 128; off > 0; off >>= 1) {
      if (t < off) {
        const float ov = red_val[t + off];
        const int   oi = red_idx[t + off];
        if (ov > red_val[t] || (ov == red_val[t] && oi < red_idx[t])) {
          red_val[t] = ov;
          red_idx[t] = oi;
        }
      }
      __syncthreads();
    }
    if (t == 0) {
      const int bi = red_idx[0];
      selmask[bi >> 5] |= (1u << (bi & 31));
      score[bi] = -INFINITY;  // exclude from later rounds
    }
    __syncthreads();
  }

  // masked rewrite: each element touched by exactly one thread (no races)
  for (int j = t; j < NN; j += 256) {
    const float val = orow[j];
    const bool sel = (selmask[j >> 5] >> (j & 31)) & 1u;
    orow[j] = sel ? val : 0.0f;
  }
}

// ---------------------------------------------------------------------------
// Inputs (setup_inputs order):
//   0: idx   int32  [4096]
//   1: emb1  f32    [4096,40]
//   2: emb2  f32    [4096,40]
//   3: W1    f32    [40,40]
//   4: b1    f32    [40]
//   5: W2    f32    [40,40]
//   6: b2    f32    [40]
//   7: noise f32    [4096,4096]
//   8: k     int32  scalar
// Output: f32 [4096,4096]
// Workspace: n1 (655KB) + n2 (655KB)
// ---------------------------------------------------------------------------
extern "C" void kernel_launch(void* const* d_in, const int* in_sizes, int n_in,
                              void* d_out, int out_size, void* d_ws,
                              size_t ws_size, hipStream_t stream) {
  (void)in_sizes; (void)n_in; (void)out_size; (void)ws_size;
  const int*   idx   = (const int*)d_in[0];
  const float* emb1  = (const float*)d_in[1];
  const float* emb2  = (const float*)d_in[2];
  const float* W1    = (const float*)d_in[3];
  const float* b1    = (const float*)d_in[4];
  const float* W2    = (const float*)d_in[5];
  const float* b2    = (const float*)d_in[6];
  const float* noise = (const float*)d_in[7];
  const int*   kptr  = (const int*)d_in[8];
  float* out = (float*)d_out;

  float* n1 = (float*)d_ws;
  float* n2 = n1 + (size_t)NN * DD;

  embed_kernel<<<(NN * DD + 255) / 256, 256, 0, stream>>>(
      idx, emb1, emb2, W1, b1, W2, b2, n1, n2);

  dim3 grid(NN / 16, NN / 16);
  adj_wmma_kernel<<<grid, 32, 0, stream>>>(n1, n2, out);

  topk_kernel<<<NN, 256, 0, stream>>>(out, noise, kptr);
}